// ScatterCFGEncodedNGramsToCFGNodeEncodings_41987600285774
// MI455X (gfx1250) — compile-verified
//
#include <hip/hip_runtime.h>

#define D 256

typedef __attribute__((ext_vector_type(2))) float v2f;
typedef __attribute__((ext_vector_type(4))) float v4f;
typedef __attribute__((ext_vector_type(8))) float v8f;

// ---------------- helpers ----------------

// no-return global fp32 atomic add at device scope (single L2 op, no CAS loop)
__device__ __forceinline__ void gfadd(float* addr, float v) {
  asm volatile("global_atomic_add_f32 %0, %1, off scope:SCOPE_DEV"
               :: "v"(addr), "v"(v) : "memory");
}

// async global -> LDS DMA, 16B per lane, tracked by ASYNCcnt (CDNA5)
__device__ __forceinline__ void async_copy_b128(unsigned lds_off, const float* gaddr) {
  asm volatile("global_load_async_to_lds_b128 %0, %1, off"
               :: "v"(lds_off), "v"(gaddr) : "memory");
}
__device__ __forceinline__ void wait_async0() {
  asm volatile("s_wait_asynccnt 0x0" ::: "memory");
}

// order-preserving float -> u32 key (monotonic under unsigned compare)
__device__ __forceinline__ unsigned fkey(float f) {
  unsigned u = __float_as_uint(f);
  return (u & 0x80000000u) ? ~u : (u | 0x80000000u);
}
__device__ __forceinline__ float funkey(unsigned k) {
  unsigned u = (k & 0x80000000u) ? (k ^ 0x80000000u) : ~k;
  return __uint_as_float(u);
}

__device__ __forceinline__ float waveReduceSum(float v) {
  // wave32 on gfx1250
  for (int off = 16; off > 0; off >>= 1) v += __shfl_xor(v, off, 32);
  return v;
}

// ---------------- kernels ----------------

// init segment-max keys to key(-inf)
__global__ void k_init_segkey(unsigned* __restrict__ segkey, int N) {
  int n = blockIdx.x * blockDim.x + threadIdx.x;
  if (n < N) segkey[n] = fkey(-__builtin_inff());
}

// cvec[n] = dot(b_k, prev[n])   (one wave per node)
__global__ void k_cvec(const float* __restrict__ prev, const float* __restrict__ bk,
                       float* __restrict__ cvec, int N) {
  int wave = (blockIdx.x * blockDim.x + threadIdx.x) >> 5;
  int lane = threadIdx.x & 31;
  if (wave >= N) return;
  const float* pr = prev + (size_t)wave * D;
  float acc = 0.f;
  for (int j = lane; j < D; j += 32) acc += pr[j] * bk[j];
  acc = waveReduceSum(acc);
  if (lane == 0) cvec[wave] = acc;
}

// T = prev @ W_k^T  via V_WMMA_F32_16X16X4_F32.  T[n][c] = sum_j prev[n][j]*Wk[c][j]
// block: 32 node rows (2 stripes), 8 waves x 2 col-tiles; each wave keeps two
// accumulators so every B tile fetched from L2 feeds two WMMAs.
__global__ void k_qk_gemm(const float* __restrict__ prev,
                          const float* __restrict__ Wk,
                          float* __restrict__ T, int N) {
  __shared__ float As[32 * 260];           // padded stride 260 -> conflict-free b64 reads
  const int n0  = blockIdx.x * 32;
  const int tid = threadIdx.x;

  // async DMA the 32xD A-stripe into LDS (16B per lane per op)
  for (int i = tid * 4; i < 32 * D; i += 256 * 4) {
    int r = i >> 8, c = i & 255;
    int n = n0 + r; if (n >= N) n = N - 1;
    async_copy_b128((unsigned)(uintptr_t)&As[r * 260 + c],
                    prev + (size_t)n * D + c);
  }
  wait_async0();
  __syncthreads();

  const int wave = tid >> 5, lane = tid & 31;
  const int lm = lane & 15, hi = lane >> 4;

  for (int t = 0; t < 2; ++t) {
    const int c0 = (wave * 2 + t) * 16;
    v8f acc0 = {0.f,0.f,0.f,0.f,0.f,0.f,0.f,0.f};
    v8f acc1 = {0.f,0.f,0.f,0.f,0.f,0.f,0.f,0.f};
    for (int k0 = 0; k0 < D; k0 += 4) {
      const int kk = k0 + 2 * hi;
      const float* bp = &Wk[(size_t)(c0 + lm) * D + kk];   // Wk[c][k], k consecutive
      v2f b = {bp[0], bp[1]};
      const float* a0p = &As[lm * 260 + kk];
      const float* a1p = &As[(16 + lm) * 260 + kk];
      v2f a0 = {a0p[0], a0p[1]};
      v2f a1 = {a1p[0], a1p[1]};
      acc0 = __builtin_amdgcn_wmma_f32_16x16x4_f32(false, a0, false, b,
                                                   (short)0, acc0, false, false);
      acc1 = __builtin_amdgcn_wmma_f32_16x16x4_f32(false, a1, false, b,
                                                   (short)0, acc1, false, false);
    }
    const int col = c0 + lm;
    for (int r = 0; r < 8; ++r) {
      int m = hi ? (r + 8) : r;
      int n = n0 + m;
      if (n < N) T[(size_t)n * D + col] = acc0[r];
      n = n0 + 16 + m;
      if (n < N) T[(size_t)n * D + col] = acc1[r];
    }
  }
}

// scores[e] = (x_e . T[idx_e] + cvec[idx_e]) / 16 ; atomic segment max.
// X is streamed non-temporally (512MB, zero reuse); T stays L2-resident.
__global__ void k_scores(const float* __restrict__ X, const float* __restrict__ T,
                         const float* __restrict__ cvec, const int* __restrict__ idx,
                         float* __restrict__ scores, unsigned* __restrict__ segkey, int E) {
  int e = (blockIdx.x * blockDim.x + threadIdx.x) >> 5;
  int lane = threadIdx.x & 31;
  if (e >= E) return;
  int n = idx[e];
  const v4f* xp = (const v4f*)(X + (size_t)e * D);
  const v4f* tp = (const v4f*)(T + (size_t)n * D);
  float acc = 0.f;
  for (int j = lane; j < D / 4; j += 32) {
    v4f a = __builtin_nontemporal_load(xp + j);
    v4f b = tp[j];
    acc += a.x * b.x + a.y * b.y + a.z * b.z + a.w * b.w;
  }
  acc = waveReduceSum(acc);
  if (lane == 0) {
    float s = (acc + cvec[n]) * 0.0625f;   // 1/sqrt(256)
    scores[e] = s;
    atomicMax(&segkey[n], fkey(s));
  }
}

// decode seg max (empty -> 0), zero denom
__global__ void k_fixmax(const unsigned* __restrict__ segkey,
                         float* __restrict__ segmax, float* __restrict__ denom, int N) {
  int n = blockIdx.x * blockDim.x + threadIdx.x;
  if (n >= N) return;
  float m = funkey(segkey[n]);
  segmax[n] = __builtin_isfinite(m) ? m : 0.0f;
  denom[n] = 0.0f;
}

// ex_e = exp(s_e - segmax); denom += ex (scores buffer rewritten in place)
__global__ void k_exp(const int* __restrict__ idx, float* __restrict__ scores,
                      const float* __restrict__ segmax, float* __restrict__ denom, int E) {
  int e = blockIdx.x * blockDim.x + threadIdx.x;
  if (e >= E) return;
  int n = idx[e];
  float ex = __expf(scores[e] - segmax[n]);
  scores[e] = ex;
  gfadd(&denom[n], ex);
}

__global__ void k_zero(float* __restrict__ p, long long cnt) {
  long long i = (long long)blockIdx.x * blockDim.x + threadIdx.x;
  if (i < cnt) p[i] = 0.0f;
}

// combined[n] += probs_e * x_e   (one wave per occurrence; L2-resident f32 atomics;
// X streamed non-temporally)
__global__ void k_combine(const float* __restrict__ X, const int* __restrict__ idx,
                          const float* __restrict__ ex, const float* __restrict__ denom,
                          float* __restrict__ comb, int E) {
  int e = (blockIdx.x * blockDim.x + threadIdx.x) >> 5;
  int lane = threadIdx.x & 31;
  if (e >= E) return;
  int n = idx[e];
  float p = ex[e] / fmaxf(denom[n], 1e-9f);
  const float* xr = X + (size_t)e * D;
  float* cr = comb + (size_t)n * D;
  for (int j = lane; j < D; j += 32) {
    float xv = __builtin_nontemporal_load(xr + j);
    gfadd(&cr[j], p * xv);
  }
}

// gate = sigmoid([prev|combined] @ W_g + b_g); out = gate*prev + (1-gate)*combined
// out holds `combined` on entry; rewritten in place (rows staged in LDS first).
__global__ void k_gate(const float* __restrict__ prev, const float* __restrict__ Wg,
                       const float* __restrict__ bg, float* __restrict__ out, int N) {
  __shared__ float As[16 * 516];           // [16][512] padded to 516
  const int n0  = blockIdx.x * 16;
  const int tid = threadIdx.x;

  // async DMA [prev | combined] stripe into LDS
  for (int i = tid * 4; i < 16 * 2 * D; i += 256 * 4) {
    int r = i >> 9, c = i & 511;
    int n = n0 + r; if (n >= N) n = N - 1;
    const float* src = (c < D) ? (prev + (size_t)n * D + c)
                               : (out  + (size_t)n * D + (c - D));
    async_copy_b128((unsigned)(uintptr_t)&As[r * 516 + c], src);
  }
  wait_async0();
  __syncthreads();

  const int wave = tid >> 5, lane = tid & 31;
  const int lm = lane & 15, hi = lane >> 4;

  for (int t = 0; t < 2; ++t) {
    const int c0 = (wave * 2 + t) * 16;
    v8f acc = {0.f,0.f,0.f,0.f,0.f,0.f,0.f,0.f};
    for (int k0 = 0; k0 < 2 * D; k0 += 4) {
      const int kk = k0 + 2 * hi;
      const float* ap = &As[lm * 516 + kk];
      v2f a = {ap[0], ap[1]};
      v2f b = {Wg[(size_t)kk * D + c0 + lm], Wg[(size_t)(kk + 1) * D + c0 + lm]};
      acc = __builtin_amdgcn_wmma_f32_16x16x4_f32(false, a, false, b,
                                                  (short)0, acc, false, false);
    }
    const int col = c0 + lm;
    const float bgv = bg[col];
    for (int r = 0; r < 8; ++r) {
      int m = hi ? (r + 8) : r;
      int n = n0 + m;
      if (n < N) {
        float g  = 1.0f / (1.0f + __expf(-(acc[r] + bgv)));
        float pv = As[m * 516 + col];
        float cv = As[m * 516 + D + col];
        out[(size_t)n * D + col] = g * pv + (1.0f - g) * cv;
      }
    }
  }
}

// ---------------- launch ----------------

extern "C" void kernel_launch(void* const* d_in, const int* in_sizes, int n_in,
                              void* d_out, int out_size, void* d_ws, size_t ws_size,
                              hipStream_t stream) {
  const float* X    = (const float*)d_in[0];
  const float* prev = (const float*)d_in[1];
  const float* Wk   = (const float*)d_in[2];
  const float* bk   = (const float*)d_in[3];
  const float* Wg   = (const float*)d_in[4];
  const float* bg   = (const float*)d_in[5];
  const int*   idx  = (const int*)d_in[6];

  const int E = in_sizes[6];
  const int N = in_sizes[1] / D;
  float* out = (float*)d_out;

  // workspace layout (floats): scores[E], cvec[N], segmax[N], denom[N], segkey[N]
  float*    scores = (float*)d_ws;
  float*    cvec   = scores + E;
  float*    segmax = cvec + N;
  float*    denom  = segmax + N;
  unsigned* segkey = (unsigned*)(denom + N);

  const int TB = 256;
  k_init_segkey<<<(N + TB - 1) / TB, TB, 0, stream>>>(segkey, N);
  k_cvec<<<(N + 7) / 8, TB, 0, stream>>>(prev, bk, cvec, N);
  k_qk_gemm<<<(N + 31) / 32, TB, 0, stream>>>(prev, Wk, out, N);           // T -> d_out
  k_scores<<<(E + 7) / 8, TB, 0, stream>>>(X, out, cvec, idx, scores, segkey, E);
  k_fixmax<<<(N + TB - 1) / TB, TB, 0, stream>>>(segkey, segmax, denom, N);
  k_exp<<<(E + TB - 1) / TB, TB, 0, stream>>>(idx, scores, segmax, denom, E);
  long long cnt = (long long)N * D;
  k_zero<<<(unsigned)((cnt + TB - 1) / TB), TB, 0, stream>>>(out, cnt);    // d_out -> combined acc
  k_combine<<<(E + 7) / 8, TB, 0, stream>>>(X, idx, scores, denom, out, E);
  k_gate<<<(N + 15) / 16, TB, 0, stream>>>(prev, Wg, bg, out, N);          // final in place
}